// GraphAttentionEncoder_22247930593628
// MI455X (gfx1250) — compile-verified
//
#include <hip/hip_runtime.h>
#include <hip/hip_bf16.h>
#include <math.h>

// ---- problem constants (match reference) ----
#define N_NODES 100000
#define IN_DIM  128
#define HID     64
#define OUT_DIM 256
#define G_DIM   16
#define N_EDGES 1600000

typedef __attribute__((ext_vector_type(2))) float v2f;
typedef __attribute__((ext_vector_type(8))) float v8f;

// ---- workspace layout (floats) ----
#define WS_H      0                        // hidden [N_NODES*HID]
#define WS_SSRC   (N_NODES * HID)          // s_src  [N_NODES]
#define WS_SDST   (WS_SSRC + N_NODES)      // s_dst  [N_NODES]
#define WS_SEGMAX (WS_SDST + N_NODES)      // segmax [N_NODES]
#define WS_DENOM  (WS_SEGMAX + N_NODES)    // denom  [N_NODES]
#define WS_POOLED (WS_DENOM + N_NODES)     // pooled [HID]   (contiguous after denom)

// ---------------------------------------------------------------------------
// K0: zero denom + pooled (contiguous region of N_NODES + HID floats)
// ---------------------------------------------------------------------------
__global__ void k_zero(float* __restrict__ p, int n) {
    int i = blockIdx.x * blockDim.x + threadIdx.x;
    if (i < n) p[i] = 0.0f;
}

// ---------------------------------------------------------------------------
// K1: hidden = elu(X @ W_in + b_in)   via V_WMMA_F32_16X16X4_F32
//   block = 256 threads = 8 waves; block tile = 32 rows x 64 cols
//   wave (w>>2) selects 16-row tile, (w&3) selects 16-col tile
//   100000 rows = 3125 blocks * 32 rows exactly -> no bounds checks,
//   EXEC stays all-ones for WMMA.
// ---------------------------------------------------------------------------
__global__ void __launch_bounds__(256)
k_gemm_elu(const float* __restrict__ X, const float* __restrict__ W,
           const float* __restrict__ bias, float* __restrict__ H) {
    const int tid  = threadIdx.x;
    const int wave = tid >> 5;
    const int lane = tid & 31;
    const int l16  = lane & 15;
    const int hi   = lane >> 4;            // 0: K={0,1}/M0-7, 1: K={2,3}/M8-15

    const int m0 = (blockIdx.x * 2 + (wave >> 2)) * 16;
    const int n0 = (wave & 3) * 16;

    // A fragment source: row (m0+l16), cols k+2*hi, k+2*hi+1
    const float* xrow = X + (size_t)(m0 + l16) * IN_DIM + 2 * hi;
    // B fragment source: W[(k+2*hi)*HID + n0+l16] and next K row
    const float* wcol = W + (size_t)(2 * hi) * HID + n0 + l16;

    v8f acc = {};
    #pragma unroll
    for (int k = 0; k < IN_DIM; k += 4) {
        float2 av = *(const float2*)(xrow + k);
        v2f a; a.x = av.x; a.y = av.y;
        v2f b; b.x = wcol[(size_t)k * HID];
               b.y = wcol[(size_t)(k + 1) * HID];
        acc = __builtin_amdgcn_wmma_f32_16x16x4_f32(
                  false, a, false, b, (short)0, acc, false, false);
    }

    const int col = n0 + l16;
    const float bv = bias[col];
    #pragma unroll
    for (int r = 0; r < 8; ++r) {
        int m = m0 + r + 8 * hi;           // C/D layout: lanes16-31 hold M=8..15
        float v = acc[r] + bv;
        H[(size_t)m * HID + col] = (v > 0.0f) ? v : (expm1f(v)); // elu, alpha=1
    }
}

// ---------------------------------------------------------------------------
// K2: per-node attention scalars; wave per node.
//   s_src[n] = h[n] . w_attn[0:64];  s_dst[n] = h[n] . w_attn[64:128]
//   segmax[n] initialized to the node's self-loop score.
// ---------------------------------------------------------------------------
__global__ void k_node_scores(const float* __restrict__ H,
                              const float* __restrict__ w_attn,
                              const float* __restrict__ b_attn,
                              float* __restrict__ s_src, float* __restrict__ s_dst,
                              float* __restrict__ segmax, int n) {
    int wid  = (blockIdx.x * blockDim.x + threadIdx.x) >> 5;
    int lane = threadIdx.x & 31;
    if (wid >= n) return;
    float2 h  = ((const float2*)(H + (size_t)wid * HID))[lane];
    float2 wa = ((const float2*)(w_attn))[lane];
    float2 wb = ((const float2*)(w_attn + HID))[lane];
    float a = h.x * wa.x + h.y * wa.y;
    float d = h.x * wb.x + h.y * wb.y;
    #pragma unroll
    for (int off = 16; off; off >>= 1) {
        a += __shfl_xor(a, off, 32);
        d += __shfl_xor(d, off, 32);
    }
    if (lane == 0) {
        s_src[wid]  = a;
        s_dst[wid]  = d;
        segmax[wid] = a + d + b_attn[0];   // self-loop edge seeds the max
    }
}

// float atomic max via monotone int punning
__device__ __forceinline__ void atomicMaxFloat(float* addr, float val) {
    if (val >= 0.0f) atomicMax((int*)addr, __float_as_int(val));
    else             atomicMin((unsigned int*)addr, (unsigned int)__float_as_int(val));
}

// ---------------------------------------------------------------------------
// K3: segment max over real edges (self-loops already seeded)
// ---------------------------------------------------------------------------
__global__ void k_edge_max(const int* __restrict__ ei,
                           const float* __restrict__ s_src,
                           const float* __restrict__ s_dst,
                           const float* __restrict__ b_attn,
                           float* __restrict__ segmax) {
    float bA = b_attn[0];
    for (int e = blockIdx.x * blockDim.x + threadIdx.x; e < N_EDGES;
         e += gridDim.x * blockDim.x) {
        int s = ei[e];
        int d = ei[N_EDGES + e];
        atomicMaxFloat(&segmax[d], s_src[s] + s_dst[d] + bA);
    }
}

// ---------------------------------------------------------------------------
// K4: denom[d] = sum_e exp(score_e - segmax[d])  over edges + self-loops
// ---------------------------------------------------------------------------
__global__ void k_edge_denom(const int* __restrict__ ei,
                             const float* __restrict__ s_src,
                             const float* __restrict__ s_dst,
                             const float* __restrict__ b_attn,
                             const float* __restrict__ segmax,
                             float* __restrict__ denom) {
    const int total = N_EDGES + N_NODES;
    float bA = b_attn[0];
    for (int idx = blockIdx.x * blockDim.x + threadIdx.x; idx < total;
         idx += gridDim.x * blockDim.x) {
        int s, d;
        if (idx < N_EDGES) { s = ei[idx]; d = ei[N_EDGES + idx]; }
        else               { s = d = idx - N_EDGES; }
        float sc = s_src[s] + s_dst[d] + bA;
        atomicAdd(&denom[d], expf(sc - segmax[d]));
    }
}

// ---------------------------------------------------------------------------
// K5: pooled[h] = sum_e attn_e * hidden[src_e][h]   (wave per edge)
//   Each lane owns 2 columns; block-level LDS reduction, then 64 atomics.
// ---------------------------------------------------------------------------
__global__ void __launch_bounds__(256)
k_edge_msg(const int* __restrict__ ei, const float* __restrict__ H,
           const float* __restrict__ s_src, const float* __restrict__ s_dst,
           const float* __restrict__ b_attn, const float* __restrict__ segmax,
           const float* __restrict__ denom, float* __restrict__ pooled) {
    __shared__ float red[HID];
    const int lane  = threadIdx.x & 31;
    const int wavesPerBlock = blockDim.x >> 5;
    const int wid    = blockIdx.x * wavesPerBlock + (threadIdx.x >> 5);
    const int nwaves = gridDim.x * wavesPerBlock;
    const int total  = N_EDGES + N_NODES;
    const float bA = b_attn[0];

    float ax = 0.0f, ay = 0.0f;
    for (int idx = wid; idx < total; idx += nwaves) {
        int s, d;
        if (idx < N_EDGES) { s = ei[idx]; d = ei[N_EDGES + idx]; }
        else               { s = d = idx - N_EDGES; }
        float sc = s_src[s] + s_dst[d] + bA;
        float w  = expf(sc - segmax[d]) / denom[d];
        float2 h = ((const float2*)(H + (size_t)s * HID))[lane];
        ax += w * h.x;
        ay += w * h.y;
    }
    if (threadIdx.x < HID) red[threadIdx.x] = 0.0f;
    __syncthreads();
    atomicAdd(&red[lane * 2],     ax);
    atomicAdd(&red[lane * 2 + 1], ay);
    __syncthreads();
    if (threadIdx.x < HID) atomicAdd(&pooled[threadIdx.x], red[threadIdx.x]);
}

// ---------------------------------------------------------------------------
// K6: out = concat(pooled/N, gf) @ W_out + b_out    (one block, 256 threads)
// ---------------------------------------------------------------------------
__global__ void k_out(const float* __restrict__ pooled,
                      const float* __restrict__ gf,
                      const float* __restrict__ W_out,
                      const float* __restrict__ b_out,
                      float* __restrict__ out) {
    int o = threadIdx.x;
    const float invN = 1.0f / (float)N_NODES;
    float acc = b_out[o];
    #pragma unroll
    for (int h = 0; h < HID; ++h)
        acc += (pooled[h] * invN) * W_out[(size_t)h * OUT_DIM + o];
    #pragma unroll
    for (int g = 0; g < G_DIM; ++g)
        acc += gf[g] * W_out[(size_t)(HID + g) * OUT_DIM + o];
    out[o] = acc;
}

// ---------------------------------------------------------------------------
extern "C" void kernel_launch(void* const* d_in, const int* in_sizes, int n_in,
                              void* d_out, int out_size, void* d_ws, size_t ws_size,
                              hipStream_t stream) {
    const float* X      = (const float*)d_in[0];   // node_features [N,128]
    const int*   ei     = (const int*)  d_in[1];   // edge_index    [2,E]
    const float* gf     = (const float*)d_in[2];   // global_features [16]
    const float* W_in   = (const float*)d_in[3];   // [128,64]
    const float* b_in   = (const float*)d_in[4];   // [64]
    const float* w_attn = (const float*)d_in[5];   // [128]
    const float* b_attn = (const float*)d_in[6];   // [1]
    const float* W_out  = (const float*)d_in[7];   // [80,256]
    const float* b_out  = (const float*)d_in[8];   // [256]
    float* out = (float*)d_out;

    float* ws = (float*)d_ws;
    float* H      = ws + WS_H;
    float* s_src  = ws + WS_SSRC;
    float* s_dst  = ws + WS_SDST;
    float* segmax = ws + WS_SEGMAX;
    float* denom  = ws + WS_DENOM;
    float* pooled = ws + WS_POOLED;

    // K0: zero denom + pooled (contiguous)
    {
        int n = N_NODES + HID;
        k_zero<<<(n + 255) / 256, 256, 0, stream>>>(denom, n);
    }
    // K1: WMMA GEMM + ELU  (100000 rows / 32 per block = 3125 blocks exactly)
    k_gemm_elu<<<N_NODES / 32, 256, 0, stream>>>(X, W_in, b_in, H);
    // K2: per-node scalars (wave per node; 8 waves/block)
    k_node_scores<<<(N_NODES + 7) / 8, 256, 0, stream>>>(H, w_attn, b_attn,
                                                         s_src, s_dst, segmax, N_NODES);
    // K3: segment max over edges
    k_edge_max<<<4096, 256, 0, stream>>>(ei, s_src, s_dst, b_attn, segmax);
    // K4: softmax denominators
    k_edge_denom<<<4096, 256, 0, stream>>>(ei, s_src, s_dst, b_attn, segmax, denom);
    // K5: pooled message accumulation (wave per edge)
    k_edge_msg<<<2048, 256, 0, stream>>>(ei, H, s_src, s_dst, b_attn,
                                         segmax, denom, pooled);
    // K6: output projection
    k_out<<<1, OUT_DIM, 0, stream>>>(pooled, gf, W_out, b_out, out);
}